// Renderer_721554506104
// MI455X (gfx1250) — compile-verified
//
#include <hip/hip_runtime.h>
#include <math.h>

typedef __attribute__((ext_vector_type(2))) float v2f;
typedef __attribute__((ext_vector_type(4))) float v4f;
typedef __attribute__((ext_vector_type(8))) float v8f;

#define PI_F   3.14159265358979323846f
#define F0_C   0.91f
#define NP     8192
#define NL     512
#define PT     65536

// ---- raw hardware transcendentals (single TRANS32 ops, no fixup code) ----
__device__ __forceinline__ float frcp(float x)  { return __builtin_amdgcn_rcpf(x); }
__device__ __forceinline__ float frsq(float x)  { return __builtin_amdgcn_rsqf(x); }
__device__ __forceinline__ float fsqrt(float x) { return __builtin_amdgcn_sqrtf(x); }

__device__ __forceinline__ float sdiv(float n, float d) {
    return fabsf(d) > 1e-12f ? n * frcp(d) : 0.0f;
}

// acos via A&S 4.4.45 polynomial (err ~7e-5): raw v_sqrt + 3 fma + select
__device__ __forceinline__ float facos(float z) {
    float ax = fabsf(z);
    float p  = fmaf(ax, fmaf(ax, fmaf(ax, -0.0187293f, 0.0742610f), -0.2121144f),
                    1.5707288f);
    float r  = fsqrt(1.0f - ax) * p;
    return z < 0.0f ? PI_F - r : r;
}

// atan2 via octant reduction + odd minimax poly; one v_rcp, branchless
__device__ __forceinline__ float fatan2(float y, float x) {
    float ax = fabsf(x), ay = fabsf(y);
    float mx = fmaxf(fmaxf(ax, ay), 1e-20f);
    float mn = fminf(ax, ay);
    float t  = mn * frcp(mx);
    float s  = t * t;
    float r  = fmaf(fmaf(fmaf(s, -0.0464964749f, 0.15931422f), s, -0.327622764f),
                    s * t, t);
    r = (ay > ax)  ? 1.57079637f - r : r;
    r = (x < 0.0f) ? 3.14159274f - r : r;
    return copysignf(r, y);
}

__device__ __forceinline__ float lin2srgb(float x) {
    x = fminf(fmaxf(x, 0.0f), 1.0f);
    float p = __builtin_amdgcn_exp2f(
                  __builtin_amdgcn_logf(fmaxf(x, 1e-8f)) * (1.0f / 2.4f));
    return x <= 0.0031308f ? 12.92f * x : fmaf(1.055f, p, -0.055f);
}

// out[t,c] = grd_rgb[t,c] * acc[t]  (the scatter==0 case of the merge)
__global__ void fill_kernel(const float* __restrict__ grd_rgb,
                            const float* __restrict__ grd_acc,
                            float* __restrict__ out) {
    int i = blockIdx.x * blockDim.x + threadIdx.x;
    if (i < PT * 3) out[i] = grd_rgb[i] * grd_acc[i / 3];
}

// One wave = 16 points (matrix column N = point = lane%16).
// Per 16-light chunk one v_wmma_f32_16x16x4_f32 computes area*(ldir.n) for
// all 16 lights x 16 points (area folded into the A matrix); lane L's D VGPR
// r is (light = chunk*16+8*(L/16)+r, point = L%16) -- exactly the 8-light
// sub-block that lane shades.  Six WMMAs (all-ones B) reduce lane-pair
// partials at the end.
__global__ __launch_bounds__(256)
void shade_kernel(const float* __restrict__ surf,
                  const float* __restrict__ ray_o,
                  const float* __restrict__ norm_raw,
                  const float* __restrict__ albedo,
                  const float* __restrict__ roughness,
                  const float* __restrict__ lvis,
                  const float* __restrict__ xyz,
                  const float* __restrict__ area,
                  const float* __restrict__ probe,
                  const float* __restrict__ grd_rgb,
                  const float* __restrict__ grd_acc,
                  const int*   __restrict__ inds,
                  float* __restrict__ out) {
    __shared__ float s_xyz4[NL * 4];     // light position {x,y,z,0}, 16B stride
    __shared__ float s_ldA[NL * 4];      // {area*ldir.x, .y, .z, 0}: A-operand slots
    __shared__ float s_p4[NL * 4];       // probe texel {r,g,b,0} for b128 taps

    const int tid = threadIdx.x;

    for (int l = tid; l < NL; l += 256) {
        float x = xyz[l * 3 + 0];
        float y = xyz[l * 3 + 1];
        float z = xyz[l * 3 + 2];
        v4f xv; xv[0] = x; xv[1] = y; xv[2] = z; xv[3] = 0.0f;
        *(v4f*)(&s_xyz4[l * 4]) = xv;                   // one ds_store_b128
        float sc = area[l] * frsq(x * x + y * y + z * z);
        v4f av; av[0] = x * sc; av[1] = y * sc; av[2] = z * sc; av[3] = 0.0f;
        *(v4f*)(&s_ldA[l * 4]) = av;                    // one ds_store_b128
        v4f pv; pv[0] = probe[l * 3 + 0]; pv[1] = probe[l * 3 + 1];
        pv[2] = probe[l * 3 + 2]; pv[3] = 0.0f;
        *(v4f*)(&s_p4[l * 4]) = pv;                     // one ds_store_b128
    }
    __syncthreads();

    const int wave  = tid >> 5;
    const int lane  = tid & 31;
    const int m     = lane & 15;   // point within wave == matrix column
    const int g     = lane >> 4;   // half: K-slots for A/B, row-half for D
    const int pbase = (blockIdx.x * 8 + wave) * 16;
    const int p     = pbase + m;

    // ---- per-point precompute ----
    float sx = surf[p * 3 + 0], sy = surf[p * 3 + 1], sz = surf[p * 3 + 2];

    float nx = norm_raw[p * 3 + 0], ny = norm_raw[p * 3 + 1], nz = norm_raw[p * 3 + 2];
    {
        float inv = frsq(fmaxf(nx * nx + ny * ny + nz * nz, 1e-20f));
        nx *= inv; ny *= inv; nz *= inv;
    }
    float vx = ray_o[p * 3 + 0] - sx, vy = ray_o[p * 3 + 1] - sy, vz = ray_o[p * 3 + 2] - sz;
    {
        float inv = frsq(fmaxf(vx * vx + vy * vy + vz * vz, 1e-20f));
        vx *= inv; vy *= inv; vz *= inv;
    }
    float rough = roughness[p];
    float alpha = rough * rough;
    float a2    = alpha * alpha;
    float a2m1  = a2 - 1.0f;

    float cos_v = nx * vx + ny * vy + nz * vz;
    float cvs   = fminf(cos_v * cos_v, 1.0f);
    float tvs   = fminf(fmaxf(sdiv(1.0f - cvs, cvs), 0.0f), 1e10f);
    float acv   = fabsf(cos_v);
    float g2v   = (acv > 1e-12f)
                    ? 2.0f * frcp(1.0f + fsqrt(1.0f + a2 * tvs)) : 0.0f;

    // B operand for the ldot WMMA: normal components in K-slot layout
    v2f Bn; Bn[0] = g ? nz : nx; Bn[1] = g ? 0.0f : ny;
    v8f z8 = {};

    float Sr = 0.f, Sg = 0.f, Sb = 0.f;   // sum of shade
    float Mr = 0.f, Mg = 0.f, Mb = 0.f;   // sum of micro*shade

    const float* lrow = lvis + p * NL;

    for (int chunk = 0; chunk < 32; ++chunk) {
        const int lbase = chunk * 16;

        // speculative prefetch of lvis two chunks ahead (dropped safely at end)
        __builtin_prefetch(lrow + lbase + 32 + 8 * g, 0, 1);

        // A operand: one unconditional ds_load_b64 (K-slot layout baked in LDS)
        v2f A = *(const v2f*)(&s_ldA[(lbase + m) * 4 + 2 * g]);

        // area*ldot for 16 lights x 16 points on the matrix pipe
        v8f dL = __builtin_amdgcn_wmma_f32_16x16x4_f32(false, A, false, Bn,
                                                       (short)0, z8, false, false);

        // this lane's 8 visibilities: two coalesced global_load_b128
        const v4f lv0 = *(const v4f*)(lrow + lbase + 8 * g);
        const v4f lv1 = *(const v4f*)(lrow + lbase + 8 * g + 4);

        #pragma unroll
        for (int r = 0; r < 8; ++r) {
            const int l = lbase + 8 * g + r;

            // light position: single aligned ds_load_b128
            const v4f xp = *(const v4f*)(&s_xyz4[l * 4]);
            float lx = xp[0] - sx;
            float ly = xp[1] - sy;
            float lz = xp[2] - sz;
            {
                float inv = frsq(lx * lx + ly * ly + lz * lz);
                lx *= inv; ly *= inv; lz *= inv;
            }
            // half-vector identities (no explicit h): |L+v|^2 = 2+2*lv
            float lv  = lx * vx + ly * vy + lz * vz;
            float ldn = lx * nx + ly * ny + lz * nz;
            float s   = frsq(fmaxf(fmaf(lv, 2.0f, 2.0f), 1e-20f));
            float ldh = (1.0f + lv) * s;          // == L.h == v.h
            float cm  = (ldn + cos_v) * s;        // == n.h
            // Fresnel
            float t1 = 1.0f - ldh;
            float t2 = t1 * t1;
            float f  = F0_C + (1.0f - F0_C) * (t2 * t2 * t1);
            // GGX D:  pi*cms^2*(a2+(1-cms)/cms)^2  ==  pi*(cms*(a2-1)+1)^2
            float cms  = cm * cm;
            float k    = fmaf(cms, a2m1, 1.0f);
            float dden = PI_F * k * k;
            float dmic = a2 * frcp(dden);
            dmic = (dden > 1e-12f) ? dmic : 0.0f;
            dmic = (cms  > 1e-12f) ? dmic : 0.0f;
            dmic = (cm   > 0.0f)   ? dmic : 0.0f;
            // Smith G view term: chv == ldh, sign test vs cos_v
            float gg = (ldh * cos_v > 0.0f) ? g2v : 0.0f;
            // specular (branchless: select only the reciprocal)
            float mden  = 4.0f * fabsf(ldn) * acv;
            float rsel  = (mden > 1e-12f) ? frcp(mden) : 0.0f;
            float micro = f * gg * dmic * rsel;

            // env-map bilinear sample of surf2light direction
            float zz    = fminf(fmaxf(lz, -1.0f + 1e-6f), 1.0f - 1e-6f);
            float theta = facos(zz);
            float phi   = fatan2(ly, lx);
            float vv = theta * (16.0f / PI_F) - 0.5f;
            float uu = fmaf(phi, 32.0f / (2.0f * PI_F), 15.5f);
            float v0 = floorf(vv), u0 = floorf(uu);
            float fv = vv - v0,  fu = uu - u0;
            int v0i = (int)v0; v0i = v0i < 0 ? 0 : (v0i > 15 ? 15 : v0i);
            int v1i = v0i + 1 > 15 ? 15 : v0i + 1;
            int u0i = ((int)u0) & 31;
            int u1i = (u0i + 1) & 31;
            // four b128 taps (texels packed {r,g,b,0})
            const v4f t00 = *(const v4f*)(&s_p4[(v0i * 32 + u0i) * 4]);
            const v4f t01 = *(const v4f*)(&s_p4[(v0i * 32 + u1i) * 4]);
            const v4f t10 = *(const v4f*)(&s_p4[(v1i * 32 + u0i) * 4]);
            const v4f t11 = *(const v4f*)(&s_p4[(v1i * 32 + u1i) * 4]);
            float w00 = (1.0f - fu) * (1.0f - fv), w01 = fu * (1.0f - fv);
            float w10 = (1.0f - fu) * fv,          w11 = fu * fv;
            float Lr = fmaf(t00[0], w00, fmaf(t01[0], w01, fmaf(t10[0], w10, t11[0] * w11)));
            float Lg = fmaf(t00[1], w00, fmaf(t01[1], w01, fmaf(t10[1], w10, t11[1] * w11)));
            float Lb = fmaf(t00[2], w00, fmaf(t01[2], w01, fmaf(t10[2], w10, t11[2] * w11)));

            float viz = (r < 4) ? lv0[r & 3] : lv1[r & 3];   // compile-time select
            float w   = viz * dL[r];                         // lvis * ldot * area

            float shr = w * Lr, shg = w * Lg, shb = w * Lb;
            Sr += shr;          Sg += shg;          Sb += shb;
            Mr += micro * shr;  Mg += micro * shg;  Mb += micro * shb;
        }
    }

    // ---- lane-pair reduction via the matrix unit (B = all ones) ----
    v2f ones; ones[0] = 1.0f; ones[1] = 1.0f;
    v2f aSr; aSr[0] = Sr; aSr[1] = 0.0f;
    v2f aSg; aSg[0] = Sg; aSg[1] = 0.0f;
    v2f aSb; aSb[0] = Sb; aSb[1] = 0.0f;
    v2f aMr; aMr[0] = Mr; aMr[1] = 0.0f;
    v2f aMg; aMg[0] = Mg; aMg[1] = 0.0f;
    v2f aMb; aMb[0] = Mb; aMb[1] = 0.0f;

    v8f dSr = __builtin_amdgcn_wmma_f32_16x16x4_f32(false, aSr, false, ones, (short)0, z8, false, false);
    v8f dSg = __builtin_amdgcn_wmma_f32_16x16x4_f32(false, aSg, false, ones, (short)0, z8, false, false);
    v8f dSb = __builtin_amdgcn_wmma_f32_16x16x4_f32(false, aSb, false, ones, (short)0, z8, false, false);
    v8f dMr = __builtin_amdgcn_wmma_f32_16x16x4_f32(false, aMr, false, ones, (short)0, z8, false, false);
    v8f dMg = __builtin_amdgcn_wmma_f32_16x16x4_f32(false, aMg, false, ones, (short)0, z8, false, false);
    v8f dMb = __builtin_amdgcn_wmma_f32_16x16x4_f32(false, aMb, false, ones, (short)0, z8, false, false);

    // D layout: lane 0 holds rows 0..7 (one per VGPR), lane 16 rows 8..15.
    if (m == 0) {
        #pragma unroll
        for (int r = 0; r < 8; ++r) {
            int row = r + 8 * g;
            int pp  = pbase + row;
            float ssr = dSr[r], ssg = dSg[r], ssb = dSb[r];
            float mmr = dMr[r], mmg = dMg[r], mmb = dMb[r];
            float ar = albedo[pp * 3 + 0] * (1.0f / PI_F);
            float ag = albedo[pp * 3 + 1] * (1.0f / PI_F);
            float ab = albedo[pp * 3 + 2] * (1.0f / PI_F);
            float rr = lin2srgb(fmaf(ar, ssr, mmr));
            float rg = lin2srgb(fmaf(ag, ssg, mmg));
            float rb = lin2srgb(fmaf(ab, ssb, mmb));
            int   t  = inds[pp];
            float ac = grd_acc[t];
            out[t * 3 + 0] = fmaf(grd_rgb[t * 3 + 0], ac, rr * (1.0f - ac));
            out[t * 3 + 1] = fmaf(grd_rgb[t * 3 + 1], ac, rg * (1.0f - ac));
            out[t * 3 + 2] = fmaf(grd_rgb[t * 3 + 2], ac, rb * (1.0f - ac));
        }
    }
}

extern "C" void kernel_launch(void* const* d_in, const int* in_sizes, int n_in,
                              void* d_out, int out_size, void* d_ws, size_t ws_size,
                              hipStream_t stream) {
    const float* surf      = (const float*)d_in[0];
    const float* ray_o     = (const float*)d_in[1];
    const float* norm_raw  = (const float*)d_in[2];
    const float* albedo    = (const float*)d_in[3];
    const float* roughness = (const float*)d_in[4];
    const float* lvis      = (const float*)d_in[5];
    const float* xyz       = (const float*)d_in[6];
    const float* area      = (const float*)d_in[7];
    const float* probe     = (const float*)d_in[8];
    const float* grd_rgb   = (const float*)d_in[9];
    const float* grd_acc   = (const float*)d_in[10];
    const int*   inds      = (const int*)d_in[11];
    float* out = (float*)d_out;

    fill_kernel<<<(PT * 3 + 255) / 256, 256, 0, stream>>>(grd_rgb, grd_acc, out);

    shade_kernel<<<NP / 128, 256, 0, stream>>>(surf, ray_o, norm_raw, albedo,
                                               roughness, lvis, xyz, area, probe,
                                               grd_rgb, grd_acc, inds, out);
    (void)d_ws; (void)ws_size; (void)n_in; (void)in_sizes; (void)out_size;
}